// LabelSmoothingLoss_7241314861302
// MI455X (gfx1250) — compile-verified
//
#include <hip/hip_runtime.h>
#include <math.h>

typedef float v2f __attribute__((ext_vector_type(2)));
typedef float v4f __attribute__((ext_vector_type(4)));
typedef float v8f __attribute__((ext_vector_type(8)));

#define THREADS 256
#define PAD_IDX 0

// Full 32-lane (wave32) sum via the matrix pipe.
// A (16x4 f32) layout: lane L<16 holds (M=L, K=0),(M=L, K=1); lane L>=16 holds
// (M=L-16, K=2),(M=L-16, K=3). Put the partial in the first slot, 0 in the
// second. B = all-ones 4x16  =>  D[m][n] = p[m] + p[m+16] for every n.
// D (16x16 f32) layout: component r of lane L holds s_{r + 8*(L/16)}, so the
// 8-component sum gives sum(s_0..s_7) in lanes 0-15 and sum(s_8..s_15) in lanes
// 16-31; one xor-16 shuffle completes the reduction.
__device__ __forceinline__ float wave_reduce_wmma(float p) {
    v2f a; a[0] = p;    a[1] = 0.0f;
    v2f b; b[0] = 1.0f; b[1] = 1.0f;
    v8f c = {0.f, 0.f, 0.f, 0.f, 0.f, 0.f, 0.f, 0.f};
    v8f d = __builtin_amdgcn_wmma_f32_16x16x4_f32(
        /*neg_a=*/false, a, /*neg_b=*/false, b,
        /*c_mod=*/(short)0, c, /*reuse_a=*/false, /*reuse_b=*/false);
    float t = ((d[0] + d[1]) + (d[2] + d[3])) + ((d[4] + d[5]) + (d[6] + d[7]));
    t += __shfl_xor(t, 16, 32);
    return t; // full wave sum, present in all lanes
}

// One workgroup per row: stream the 50257-float row with coalesced NT b128
// loads, reduce, and emit the closed-form per-row loss into row_out[row].
// conf/sv/cent are problem constants, computed on the host in f64.
__global__ __launch_bounds__(THREADS)
void ls_row_kernel(const float* __restrict__ x, const int* __restrict__ tgt,
                   float* __restrict__ row_out, int V,
                   float conf, float sv, float cent) {
    const int row = blockIdx.x;
    const int tid = threadIdx.x;
    const float* rp = x + (size_t)row * (size_t)V;

    // Rows are only 4B-aligned (V*4 % 16 == 4): scalar prologue to 16B.
    const int mis = (int)((((unsigned long long)rp) >> 2) & 3ull);
    const int pro = (4 - mis) & 3;

    float acc0 = 0.0f, acc1 = 0.0f;
    if (tid < pro) acc0 += rp[tid];

    const v4f* vp = (const v4f*)(rp + pro);
    const int nvec = (V - pro) >> 2;
    const int rem  = (V - pro) & 3;

    int i = tid;
    for (; i + THREADS < nvec; i += 2 * THREADS) {
        v4f t0 = __builtin_nontemporal_load(&vp[i]);
        v4f t1 = __builtin_nontemporal_load(&vp[i + THREADS]);
        acc0 += (t0[0] + t0[1]) + (t0[2] + t0[3]);
        acc1 += (t1[0] + t1[1]) + (t1[2] + t1[3]);
    }
    if (i < nvec) {
        v4f t0 = __builtin_nontemporal_load(&vp[i]);
        acc0 += (t0[0] + t0[1]) + (t0[2] + t0[3]);
    }
    if (tid < rem) acc1 += rp[pro + nvec * 4 + tid];

    const float w = wave_reduce_wmma(acc0 + acc1);

    __shared__ float lds[THREADS / 32];
    if ((tid & 31) == 0) lds[tid >> 5] = w;
    __syncthreads();

    if (tid == 0) {
        float rowsum = 0.0f;
#pragma unroll
        for (int k = 0; k < THREADS / 32; ++k) rowsum += lds[k];
        const int t = tgt[row];
        float res = 0.0f;
        if (t != PAD_IDX) {
            const float x0 = rp[0];
            const float xt = rp[t];
            res = cent - conf * xt - sv * (rowsum - x0 - xt);
        }
        row_out[row] = res;
    }
}

// Deterministic final reduction of the N per-row results (no float atomics, so
// graph replays are bit-stable).
__global__ __launch_bounds__(THREADS)
void ls_final_kernel(const float* __restrict__ row_out, float* __restrict__ out,
                     int n) {
    const int tid = threadIdx.x;
    float acc = 0.0f;
    for (int i = tid; i < n; i += THREADS) acc += row_out[i];
    const float w = wave_reduce_wmma(acc);
    __shared__ float lds[THREADS / 32];
    if ((tid & 31) == 0) lds[tid >> 5] = w;
    __syncthreads();
    if (tid == 0) {
        float s = 0.0f;
#pragma unroll
        for (int k = 0; k < THREADS / 32; ++k) s += lds[k];
        out[0] = s;
    }
}

extern "C" void kernel_launch(void* const* d_in, const int* in_sizes, int n_in,
                              void* d_out, int out_size, void* d_ws, size_t ws_size,
                              hipStream_t stream) {
    const float* x   = (const float*)d_in[0];  // [N, V] fp32 log-probs
    const int*   tgt = (const int*)d_in[1];    // [N] targets
    float* out = (float*)d_out;                // scalar loss
    float* ws  = (float*)d_ws;                 // N floats of scratch

    const int N = in_sizes[1];
    const int V = in_sizes[0] / N;

    // Problem constants in f64 on the host:
    //   sv   = smoothing/(V-2)
    //   cent = conf*ln(conf) + (V-2)*sv*ln(sv)   (per-row entropy term)
    const double smoothing = 0.1;
    const double confd = 1.0 - smoothing;
    const double svd   = smoothing / (double)(V - 2);
    const double centd = confd * log(confd) + smoothing * log(svd);

    ls_row_kernel<<<N, THREADS, 0, stream>>>(x, tgt, ws, V,
                                             (float)confd, (float)svd,
                                             (float)centd);
    ls_final_kernel<<<1, THREADS, 0, stream>>>(ws, out, N);
}